// MultiHeadAttentionPooling_16466904613541
// MI455X (gfx1250) — compile-verified
//
#include <hip/hip_runtime.h>
#include <hip/hip_bf16.h>
#include <stdint.h>

typedef __attribute__((ext_vector_type(2))) float v2f;
typedef __attribute__((ext_vector_type(8))) float v8f;

#define HEADS  8
#define CH     32
#define CHUNK  64
#define KSTEPS (CHUNK / 4)

// ---------------- zero the output (harness poisons it) ----------------
__global__ void __launch_bounds__(256) zero_out_kernel(float* out, int n) {
    int i = blockIdx.x * blockDim.x + threadIdx.x;
    if (i < n) out[i] = 0.0f;
}

// ---------------- per-head softmax over M slots -> w[H][M] ----------------
__global__ void __launch_bounds__(256) softmax_rows_kernel(const float* __restrict__ att,
                                                           float* __restrict__ w, int M) {
    __shared__ float red[256];
    const int h   = blockIdx.x;
    const int tid = threadIdx.x;
    const float* row  = att + (size_t)h * M;
    float*       wrow = w   + (size_t)h * M;

    float mx = -3.402823466e38f;
    for (int i = tid; i < M; i += 256) mx = fmaxf(mx, row[i]);
    red[tid] = mx; __syncthreads();
    for (int s = 128; s > 0; s >>= 1) {
        if (tid < s) red[tid] = fmaxf(red[tid], red[tid + s]);
        __syncthreads();
    }
    mx = red[0]; __syncthreads();

    float sum = 0.0f;
    for (int i = tid; i < M; i += 256) {
        float e = __expf(row[i] - mx);
        wrow[i] = e;
        sum += e;
    }
    red[tid] = sum; __syncthreads();
    for (int s = 128; s > 0; s >>= 1) {
        if (tid < s) red[tid] += red[tid + s];
        __syncthreads();
    }
    const float inv = 1.0f / red[0];
    for (int i = tid; i < M; i += 256) wrow[i] *= inv;
}

// ---------------- WMMA weighted segment-sum pooling ----------------
// One wave per 64-node chunk: pooled[8x32] = Wmask[8x64] @ X[64x32] via
// v_wmma_f32_16x16x4_f32 (16 K-steps x 2 channel tiles).
__global__ void __launch_bounds__(32) wmma_pool_kernel(const float* __restrict__ x,
                                                       const long long* __restrict__ batch,
                                                       const float* __restrict__ w,
                                                       float* __restrict__ out,
                                                       int N, int M) {
    __shared__ float xs[CHUNK][CH];     // node-major x tile        (8 KB)
    __shared__ float wsh[16][CHUNK];    // 16 head rows, 8..15 zero (4 KB)
    __shared__ int   bs[CHUNK];         // graph ids

    const int       lane = threadIdx.x;          // 0..31 (wave32)
    const long long n0   = (long long)blockIdx.x * CHUNK;
    const long long rem  = (long long)N - n0;
    const int  nvalid    = rem < CHUNK ? (int)rem : CHUNK;
    const bool full      = (nvalid == CHUNK);

    // prefetch next chunk of x into L2 (global_prefetch_b8)
    if (n0 + 2 * CHUNK <= (long long)N)
        __builtin_prefetch(x + (n0 + CHUNK) * CH + (long long)lane * 16, 0, 1);

    // ---- stage x tile: CDNA5 async global->LDS for full chunks ----
    if (full) {
        const uint32_t lds0 = (uint32_t)(uintptr_t)(&xs[0][0]);
        const uint64_t gb   = (uint64_t)(uintptr_t)(x + n0 * CH);
#pragma unroll
        for (int t = 0; t < (CHUNK * CH * 4) / (32 * 16); ++t) {   // 16 x B128 per lane-set
            const uint32_t laddr = lds0 + (uint32_t)((t * 32 + lane) * 16);
            const uint64_t gaddr = gb + (uint64_t)((t * 32 + lane) * 16);
            asm volatile("global_load_async_to_lds_b128 %0, %1, off"
                         :: "v"(laddr), "v"(gaddr) : "memory");
        }
    } else {
        const float4* xg = (const float4*)(x + n0 * CH);
        float4*       xl = (float4*)&xs[0][0];
        const int nv4 = nvalid * (CH / 4);
        for (int i = lane; i < CHUNK * (CH / 4); i += 32) {
            float4 v = make_float4(0.f, 0.f, 0.f, 0.f);
            if (i < nv4) v = xg[i];
            xl[i] = v;
        }
    }

    // graph ids (invalid slots -> -1, never matches g >= 0)
    for (int i = lane; i < CHUNK; i += 32)
        bs[i] = (i < nvalid) ? (int)batch[n0 + i] : -1;

    // weights, padded to 16 rows so A-fragment loads need no row predicate
    for (int i = lane; i < 16 * CHUNK; i += 32) {
        const int h = i >> 6;             // / CHUNK
        const int j = i & (CHUNK - 1);
        float v = 0.0f;
        if (h < HEADS && j < nvalid)
            v = w[(size_t)h * M + (int)((n0 + j) % (long long)M)];
        wsh[h][j] = v;
    }

    if (full) asm volatile("s_wait_asynccnt 0x0" ::: "memory");
    __syncthreads();

    const int half = lane >> 4;           // which K-pair this lane carries
    const int mrow = lane & 15;           // A row (head) / B col (channel)

    // ---- preload all fragments into registers (no LDS in the WMMA loop) ----
    float aw0[KSTEPS], aw1[KSTEPS];
    v2f   bf0[KSTEPS], bf1[KSTEPS];
#pragma unroll
    for (int k = 0; k < KSTEPS; ++k) {
        const int ka = 4 * k + 2 * half;
        aw0[k] = wsh[mrow][ka];
        aw1[k] = wsh[mrow][ka + 1];
        bf0[k].x = xs[ka][mrow];       bf0[k].y = xs[ka + 1][mrow];
        bf1[k].x = xs[ka][16 + mrow];  bf1[k].y = xs[ka + 1][16 + mrow];
    }

    const int bmin = bs[0];
    const int bmax = bs[nvalid - 1];

    if (full && bmin == bmax) {
        // ---- fast path: whole chunk in one graph, no masking ----
        v8f c0 = {}, c1 = {};
#pragma unroll
        for (int k = 0; k < KSTEPS; ++k) {
            v2f a; a.x = aw0[k]; a.y = aw1[k];
            c0 = __builtin_amdgcn_wmma_f32_16x16x4_f32(false, a, false, bf0[k],
                                                       (short)0, c0, false, false);
            c1 = __builtin_amdgcn_wmma_f32_16x16x4_f32(false, a, false, bf1[k],
                                                       (short)0, c1, false, false);
        }
        if (half == 0) {
            float* og = out + (size_t)bmin * (HEADS * CH);
#pragma unroll
            for (int r = 0; r < 8; ++r) {
                atomicAdd(&og[r * CH + mrow],      c0[r]);
                atomicAdd(&og[r * CH + 16 + mrow], c1[r]);
            }
        }
    } else {
        // ---- boundary/tail path: mask per graph id (v_cmp + v_cndmask only) ----
        int ab0[KSTEPS], ab1[KSTEPS];
#pragma unroll
        for (int k = 0; k < KSTEPS; ++k) {
            const int ka = 4 * k + 2 * half;
            ab0[k] = bs[ka];
            ab1[k] = bs[ka + 1];
        }
        for (int g = bmin; g <= bmax; ++g) {
            v8f c0 = {}, c1 = {};
#pragma unroll
            for (int k = 0; k < KSTEPS; ++k) {
                v2f a;
                a.x = (ab0[k] == g) ? aw0[k] : 0.0f;
                a.y = (ab1[k] == g) ? aw1[k] : 0.0f;
                c0 = __builtin_amdgcn_wmma_f32_16x16x4_f32(false, a, false, bf0[k],
                                                           (short)0, c0, false, false);
                c1 = __builtin_amdgcn_wmma_f32_16x16x4_f32(false, a, false, bf1[k],
                                                           (short)0, c1, false, false);
            }
            if (half == 0) {
                float* og = out + (size_t)g * (HEADS * CH);
#pragma unroll
                for (int r = 0; r < 8; ++r) {
                    atomicAdd(&og[r * CH + mrow],      c0[r]);
                    atomicAdd(&og[r * CH + 16 + mrow], c1[r]);
                }
            }
        }
    }
}

extern "C" void kernel_launch(void* const* d_in, const int* in_sizes, int n_in,
                              void* d_out, int out_size, void* d_ws, size_t ws_size,
                              hipStream_t stream) {
    const float*     x     = (const float*)d_in[0];
    const long long* batch = (const long long*)d_in[1];
    const float*     att   = (const float*)d_in[2];

    const int N = in_sizes[1];            // total nodes
    const int M = in_sizes[2] / HEADS;    // node slots per graph (softmax axis)

    float* wbuf = (float*)d_ws;           // softmax weights [HEADS][M]
    float* out  = (float*)d_out;

    zero_out_kernel<<<(out_size + 255) / 256, 256, 0, stream>>>(out, out_size);
    softmax_rows_kernel<<<HEADS, 256, 0, stream>>>(att, wbuf, M);

    const int nblocks = (N + CHUNK - 1) / CHUNK;
    wmma_pool_kernel<<<nblocks, 32, 0, stream>>>(x, batch, wbuf, out, N, M);
}